// GNN_8701603741997
// MI455X (gfx1250) — compile-verified
//
#include <hip/hip_runtime.h>
#include <hip/hip_bf16.h>
#include <math.h>

#define HID 128
#define SLOPE 0.2f

typedef __attribute__((ext_vector_type(2))) float v2f;
typedef __attribute__((ext_vector_type(8))) float v8f;

// ============================================================================
// XP = H(nNodes x 128) * W(128 x 128), fp32 via V_WMMA_F32_16X16X4_F32.
// One wave computes one 16x16 output tile; 8 waves/block cover N=128 columns
// of a 16-row strip. K=128 in steps of 4 -> 32 WMMA ops per wave.
// A 16x4 layout (ISA 7.12.2): lanes0-15 VGPR0/1 = K=k,k+1 ; lanes16-31 = k+2,k+3
// B 4x16 layout mirrors the K split; C/D: VGPR r -> rows r (lanes0-15), r+8.
// Row guard is applied ONLY at the store: an out-of-range lane merely loads a
// clamped (valid) row and pollutes output rows that are never written, so the
// K-loop is completely branch-free (no EXEC save/restore around WMMA).
// ============================================================================
__global__ __launch_bounds__(256) void gat_gemm_wmma(
    const float* __restrict__ H, const float* __restrict__ W,
    float* __restrict__ XP, int nNodes)
{
  const int wave = threadIdx.x >> 5;     // 0..7 -> column tile
  const int lane = threadIdx.x & 31;
  const int half = lane >> 4;            // 0 = lanes 0-15, 1 = lanes 16-31
  const int l    = lane & 15;
  const int m0 = blockIdx.x * 16;
  const int n0 = wave * 16;

  const int rowA = min(m0 + l, nNodes - 1);          // clamp, no branch
  const float* __restrict__ hrow = H + (size_t)rowA * HID + 2 * half;
  const float* __restrict__ wcol = W + (size_t)(2 * half) * HID + n0 + l;

  v8f c = {};
#pragma unroll
  for (int k = 0; k < HID; k += 4) {
    v2f a = *(const v2f*)(hrow + k);                 // 8B-aligned K pair
    v2f b;
    b.x = wcol[(size_t)k * HID];                     // K = k + 2*half
    b.y = wcol[(size_t)k * HID + HID];               // K = k + 2*half + 1
#if __has_builtin(__builtin_amdgcn_wmma_f32_16x16x4_f32)
    c = __builtin_amdgcn_wmma_f32_16x16x4_f32(
            /*neg_a=*/false, a, /*neg_b=*/false, b,
            /*c_mod=*/(short)0, c, /*reuse_a=*/false, /*reuse_b=*/false);
#else
    for (int r = 0; r < 8; ++r) c[r] += a.x * b.x + a.y * b.y;  // never taken
#endif
  }
#pragma unroll
  for (int r = 0; r < 8; ++r) {
    int row = m0 + r + 8 * half;
    if (row < nNodes) XP[(size_t)row * HID + n0 + l] = c[r];
  }
}

// ---------------- layer 1: rank-1 transform xp[v,c] = x[v] * W1[c] ----------
__global__ void l1_xp(const float* __restrict__ x, const float* __restrict__ W1,
                      float* __restrict__ XP, int nNodes)
{
  int idx = blockIdx.x * blockDim.x + threadIdx.x;
  if (idx >= nNodes * HID) return;
  int v = idx >> 7, ccol = idx & (HID - 1);
  XP[idx] = x[v] * W1[ccol];
}

// ---------------- attention logits: al_s[v]=xp[v]·a_src, al_d[v]=xp[v]·a_dst -
__global__ __launch_bounds__(256) void attn_logits(
    const float* __restrict__ XP, const float* __restrict__ a_src,
    const float* __restrict__ a_dst, float* __restrict__ al_s,
    float* __restrict__ al_d, int nNodes)
{
  int wave = threadIdx.x >> 5, lane = threadIdx.x & 31;
  int v = blockIdx.x * 8 + wave;
  if (v >= nNodes) return;
  const float* row = XP + (size_t)v * HID;
  float ss = 0.f, sd = 0.f;
#pragma unroll
  for (int j = 0; j < 4; ++j) {
    int cidx = lane + 32 * j;
    float xv = row[cidx];
    ss += xv * a_src[cidx];
    sd += xv * a_dst[cidx];
  }
#pragma unroll
  for (int off = 16; off; off >>= 1) {
    ss += __shfl_xor(ss, off, 32);
    sd += __shfl_xor(sd, off, 32);
  }
  if (lane == 0) { al_s[v] = ss; al_d[v] = sd; }
}

// ---------------- zero accumulators ----------------------------------------
__global__ void seg_init(float* __restrict__ S, unsigned* __restrict__ m_enc,
                         float* __restrict__ den, int nNodes)
{
  int idx = blockIdx.x * blockDim.x + threadIdx.x;
  if (idx < nNodes * HID) S[idx] = 0.f;
  if (idx < nNodes) { m_enc[idx] = 0u; den[idx] = 0.f; }
}

// monotone float<->uint encoding so atomicMax(uint) == float max (exact)
__device__ __forceinline__ unsigned enc_f(float f) {
  unsigned b = __float_as_uint(f);
  return (b & 0x80000000u) ? ~b : (b | 0x80000000u);
}
__device__ __forceinline__ float dec_f(unsigned k) {
  return (k & 0x80000000u) ? __uint_as_float(k & 0x7fffffffu)
                           : __uint_as_float(~k);
}

// ---------------- pass 1: alpha + segment max over dst ----------------------
__global__ void edge_max(const int* __restrict__ ei,
                         const float* __restrict__ al_s,
                         const float* __restrict__ al_d,
                         float* __restrict__ alpha,
                         unsigned* __restrict__ m_enc,
                         int nEdges, int nNodes)
{
  int t = blockIdx.x * blockDim.x + threadIdx.x;
  int tot = nEdges + nNodes;
  if (t >= tot) return;
  int s, d;
  if (t < nEdges) { s = ei[t]; d = ei[nEdges + t]; }
  else            { s = d = t - nEdges; }           // self loop
  float a = al_s[s] + al_d[d];
  a = (a > 0.f) ? a : SLOPE * a;                    // leaky_relu
  alpha[t] = a;
  atomicMax(m_enc + d, enc_f(a));
}

// ---------------- pass 2: e = exp(alpha - m); den += e; S[d] += e * xp[s] ---
__global__ __launch_bounds__(256) void edge_accum(
    const int* __restrict__ ei, const float* __restrict__ alpha,
    const unsigned* __restrict__ m_enc, const float* __restrict__ XP,
    float* __restrict__ S, float* __restrict__ den, int nEdges, int nNodes)
{
  int wave = threadIdx.x >> 5, lane = threadIdx.x & 31;
  int t = blockIdx.x * 8 + wave;
  int tot = nEdges + nNodes;
  if (t >= tot) return;
  int s, d;
  if (t < nEdges) { s = ei[t]; d = ei[nEdges + t]; }
  else            { s = d = t - nEdges; }
  float e = expf(alpha[t] - dec_f(m_enc[d]));
  if (lane == 0) atomicAdd(den + d, e);
  const float* xr = XP + (size_t)s * HID;
  float* sr = S + (size_t)d * HID;
#pragma unroll
  for (int j = 0; j < 4; ++j) {
    int cidx = lane + 32 * j;
    atomicAdd(sr + cidx, e * xr[cidx]);
  }
}

// ---------------- finalize: h = relu(S/den + b) -----------------------------
__global__ void finalize(const float* __restrict__ S,
                         const float* __restrict__ den,
                         const float* __restrict__ bias,
                         float* __restrict__ Hout, int nNodes)
{
  int idx = blockIdx.x * blockDim.x + threadIdx.x;
  if (idx >= nNodes * HID) return;
  int v = idx >> 7, ccol = idx & (HID - 1);
  float o = S[idx] / den[v] + bias[ccol];
  Hout[idx] = (o > 0.f) ? o : 0.f;
}

// ---------------- global mean pool + head -----------------------------------
__global__ void pool_zero(float* __restrict__ psum, float* __restrict__ pcnt,
                          int nGraphs)
{
  int idx = blockIdx.x * blockDim.x + threadIdx.x;
  if (idx < nGraphs * HID) psum[idx] = 0.f;
  if (idx < nGraphs) pcnt[idx] = 0.f;
}

__global__ void pool_acc(const float* __restrict__ H,
                         const int* __restrict__ batch,
                         float* __restrict__ psum, float* __restrict__ pcnt,
                         int nNodes)
{
  int idx = blockIdx.x * blockDim.x + threadIdx.x;
  if (idx >= nNodes * HID) return;
  int v = idx >> 7, ccol = idx & (HID - 1);
  int g = batch[v];
  atomicAdd(psum + (size_t)g * HID + ccol, H[idx]);
  if (ccol == 0) atomicAdd(pcnt + g, 1.0f);
}

__global__ void head(const float* __restrict__ psum,
                     const float* __restrict__ pcnt,
                     const float* __restrict__ Wlin,
                     const float* __restrict__ blin,
                     float* __restrict__ out, int nGraphs)
{
  int t = blockIdx.x * blockDim.x + threadIdx.x;
  if (t >= nGraphs * 2) return;
  int g = t >> 1, o = t & 1;
  float cnt = fmaxf(pcnt[g], 1.0f);
  float acc = 0.f;
  for (int c = 0; c < HID; ++c)
    acc += (psum[(size_t)g * HID + c] / cnt) * Wlin[c * 2 + o];
  acc += blin[o];
  out[t] = 1.f / (1.f + expf(-acc));
}

// ============================================================================
extern "C" void kernel_launch(void* const* d_in, const int* in_sizes, int n_in,
                              void* d_out, int out_size, void* d_ws,
                              size_t ws_size, hipStream_t stream)
{
  (void)n_in; (void)ws_size;
  const float* x     = (const float*)d_in[0];
  const int*   ei    = (const int*)d_in[1];     // [2,E] (JAX x64 off -> int32)
  const int*   batch = (const int*)d_in[2];
  const float* W1  = (const float*)d_in[3];
  const float* as1 = (const float*)d_in[4];
  const float* ad1 = (const float*)d_in[5];
  const float* b1  = (const float*)d_in[6];
  const float* W2  = (const float*)d_in[7];
  const float* as2 = (const float*)d_in[8];
  const float* ad2 = (const float*)d_in[9];
  const float* b2  = (const float*)d_in[10];
  const float* W3  = (const float*)d_in[11];
  const float* as3 = (const float*)d_in[12];
  const float* ad3 = (const float*)d_in[13];
  const float* b3  = (const float*)d_in[14];
  const float* Wlin = (const float*)d_in[15];
  const float* blin = (const float*)d_in[16];

  const int nNodes  = in_sizes[0];
  const int nEdges  = in_sizes[1] / 2;
  const int nGraphs = out_size / 2;
  const int totE    = nEdges + nNodes;          // with self loops

  // workspace layout (fp32 elements)
  float*    ws    = (float*)d_ws;
  float*    bufA  = ws;                               // [N,128] layer io
  float*    bufXP = bufA  + (size_t)nNodes * HID;     // [N,128] xp
  float*    bufS  = bufXP + (size_t)nNodes * HID;     // [N,128] accum
  float*    alpha = bufS  + (size_t)nNodes * HID;     // [E+N]
  float*    al_s  = alpha + (size_t)totE;             // [N]
  float*    al_d  = al_s  + (size_t)nNodes;           // [N]
  unsigned* m_enc = (unsigned*)(al_d + (size_t)nNodes); // [N]
  float*    den   = (float*)(m_enc + (size_t)nNodes); // [N]
  float*    psum  = den + (size_t)nNodes;             // [G,128]
  float*    pcnt  = psum + (size_t)nGraphs * HID;     // [G]

  const int TPB = 256;
  const int gNC  = (nNodes * HID + TPB - 1) / TPB;    // node*channel grid
  const int gN8  = (nNodes + 7) / 8;                  // wave-per-node grid
  const int gE   = (totE + TPB - 1) / TPB;            // thread-per-edge grid
  const int gE8  = (totE + 7) / 8;                    // wave-per-edge grid
  const int gMM  = (nNodes + 15) / 16;                // 16-row GEMM strips

  const float* Ws[3] = {W1, W2, W3};
  const float* As[3] = {as1, as2, as3};
  const float* Ad[3] = {ad1, ad2, ad3};
  const float* Bs[3] = {b1, b2, b3};

  for (int layer = 0; layer < 3; ++layer) {
    if (layer == 0)
      l1_xp<<<gNC, TPB, 0, stream>>>(x, Ws[0], bufXP, nNodes);
    else
      gat_gemm_wmma<<<gMM, TPB, 0, stream>>>(bufA, Ws[layer], bufXP, nNodes);

    attn_logits<<<gN8, TPB, 0, stream>>>(bufXP, As[layer], Ad[layer],
                                         al_s, al_d, nNodes);
    seg_init<<<gNC, TPB, 0, stream>>>(bufS, m_enc, den, nNodes);
    edge_max<<<gE, TPB, 0, stream>>>(ei, al_s, al_d, alpha, m_enc,
                                     nEdges, nNodes);
    edge_accum<<<gE8, TPB, 0, stream>>>(ei, alpha, m_enc, bufXP, bufS, den,
                                        nEdges, nNodes);
    finalize<<<gNC, TPB, 0, stream>>>(bufS, den, Bs[layer], bufA, nNodes);
  }

  pool_zero<<<(nGraphs * HID + TPB - 1) / TPB, TPB, 0, stream>>>(psum, pcnt,
                                                                 nGraphs);
  pool_acc<<<gNC, TPB, 0, stream>>>(bufA, batch, psum, pcnt, nNodes);
  head<<<(nGraphs * 2 + TPB - 1) / TPB, TPB, 0, stream>>>(
      psum, pcnt, Wlin, blin, (float*)d_out, nGraphs);
}